// CapsuleUnit_35596688949588
// MI455X (gfx1250) — compile-verified
//
#include <hip/hip_runtime.h>
#include <math.h>

typedef __attribute__((ext_vector_type(2))) float v2f;
typedef __attribute__((ext_vector_type(8))) float v8f;

#define K_DIM    512
#define N_DIM    512
#define IN_CAPS  1152
#define BATCH    64
#define M_DIM    (BATCH * IN_CAPS)   // 73728
#define N_ITERS  10
#define SJ_SPLIT 9                    // 1152 / 128 i-chunks

// ---------------------------------------------------------------------------
// One-time transpose of W [K][N] -> Wt [N][K] so B fragments become
// contiguous (k, k+1) pairs loadable with global_load_b64.
// ---------------------------------------------------------------------------
__global__ __launch_bounds__(256)
void transpose_w(const float* __restrict__ W, float* __restrict__ Wt) {
  __shared__ float tile[32][33];
  const int bx = blockIdx.x * 32;   // k base
  const int by = blockIdx.y * 32;   // n base
  const int tx = threadIdx.x & 31;
  const int ty = threadIdx.x >> 5;  // 0..7
#pragma unroll
  for (int r = 0; r < 32; r += 8)
    tile[ty + r][tx] = W[(size_t)(bx + ty + r) * N_DIM + (by + tx)];
  __syncthreads();
#pragma unroll
  for (int r = 0; r < 32; r += 8)
    Wt[(size_t)(by + ty + r) * K_DIM + (bx + tx)] = tile[tx][ty + r];
}

// ---------------------------------------------------------------------------
// GEMM: U[m,n] = sum_k A[m,k] * Wt[n,k] + bias[n]
// A = input_x viewed as [73728, 512] row-major; Wt = W transposed [N][K].
// One wave (32 lanes) per block computes a 64M x 64N macro-tile using
// V_WMMA_F32_16X16X4_F32 (16 accumulator fragments, K-step of 4).
// Fragment layouts per CDNA5 ISA 7.12.2:
//   A 16x4 f32 : lanes 0-15 -> K=0(v0),K=1(v1); lanes 16-31 -> K=2(v0),K=3(v1)
//   B 4x16 f32 : v0: lanes0-15 row K=0 / lanes16-31 row K=2; v1: K=1 / K=3
//   C/D 16x16  : VGPR r: lanes0-15 M=r, lanes16-31 M=r+8; N = lane%16
// Grid: x = N-chunk (8, fastest) so the 8 blocks sharing an A tile dispatch
// consecutively and hit L2; y = M-chunk (1152).
// ---------------------------------------------------------------------------
__global__ __launch_bounds__(32)
void caps_gemm_wmma(const float* __restrict__ A, const float* __restrict__ Wt,
                    const float* __restrict__ bias, float* __restrict__ U) {
  const int lane   = threadIdx.x;
  const int laneLo = lane & 15;
  const int laneHi = lane >> 4;            // 0 or 1
  const int koff   = 2 * laneHi;           // K sub-offset for this half-wave
  const int tileN  = blockIdx.x * 64;
  const int tileM  = blockIdx.y * 64;

  v8f acc[4][4];
#pragma unroll
  for (int mt = 0; mt < 4; ++mt)
#pragma unroll
    for (int nt = 0; nt < 4; ++nt)
      acc[mt][nt] = {};

  for (int kb = 0; kb < K_DIM; kb += 4) {
    v2f afrag[4];
#pragma unroll
    for (int mt = 0; mt < 4; ++mt) {
      const float* ap =
          A + (size_t)(tileM + mt * 16 + laneLo) * K_DIM + kb + koff;
      afrag[mt] = *(const v2f*)ap;         // 8B-aligned pair (K, K+1)
    }
    v2f bfrag[4];
#pragma unroll
    for (int nt = 0; nt < 4; ++nt) {
      const int col = tileN + nt * 16 + laneLo;
      const float* bp = Wt + (size_t)col * K_DIM + kb + koff;
      bfrag[nt] = *(const v2f*)bp;         // 8B-aligned pair (K, K+1)
    }
#pragma unroll
    for (int mt = 0; mt < 4; ++mt)
#pragma unroll
      for (int nt = 0; nt < 4; ++nt)
        acc[mt][nt] = __builtin_amdgcn_wmma_f32_16x16x4_f32(
            false, afrag[mt], false, bfrag[nt], (short)0, acc[mt][nt],
            false, false);
  }

  // Epilogue: + bias, store u.
#pragma unroll
  for (int nt = 0; nt < 4; ++nt) {
    const int col = tileN + nt * 16 + laneLo;
    const float bv = bias[col];
#pragma unroll
    for (int mt = 0; mt < 4; ++mt) {
#pragma unroll
      for (int r = 0; r < 8; ++r) {
        const int row = tileM + mt * 16 + r + 8 * laneHi;
        U[(size_t)row * N_DIM + col] = acc[mt][nt][r] + bv;
      }
    }
  }
}

// ---------------------------------------------------------------------------
// c initialization from coeffs input
// ---------------------------------------------------------------------------
__global__ void init_c(const float* __restrict__ coeffs, float* __restrict__ c) {
  int i = blockIdx.x * 256 + threadIdx.x;
  if (i < IN_CAPS) c[i] = coeffs[i];
}

// ---------------------------------------------------------------------------
// softmax over the 1152 routing logits (single block, deterministic LDS tree)
// ---------------------------------------------------------------------------
__global__ __launch_bounds__(256)
void softmax_c(const float* __restrict__ c, float* __restrict__ cij) {
  __shared__ float red[256];
  const int t = threadIdx.x;

  float m = -INFINITY;
  for (int i = t; i < IN_CAPS; i += 256) m = fmaxf(m, c[i]);
  red[t] = m;
  __syncthreads();
  for (int s = 128; s > 0; s >>= 1) {
    if (t < s) red[t] = fmaxf(red[t], red[t + s]);
    __syncthreads();
  }
  const float mx = red[0];
  __syncthreads();

  float sum = 0.f;
  for (int i = t; i < IN_CAPS; i += 256) sum += expf(c[i] - mx);
  red[t] = sum;
  __syncthreads();
  for (int s = 128; s > 0; s >>= 1) {
    if (t < s) red[t] += red[t + s];
    __syncthreads();
  }
  const float inv = 1.f / red[0];

  for (int i = t; i < IN_CAPS; i += 256) cij[i] = expf(c[i] - mx) * inv;
}

// ---------------------------------------------------------------------------
// sj partials: sjp[part][b][o] = sum_{i in chunk} u[b,i,o] * cij[i]
// Fixed split of 9 chunks of 128 capsules -> deterministic final reduce.
// ---------------------------------------------------------------------------
__global__ __launch_bounds__(256)
void sj_partial(const float* __restrict__ U, const float* __restrict__ cij,
                float* __restrict__ sjp) {
  const int b = blockIdx.x;
  const int part = blockIdx.y;
  const int t = threadIdx.x;
  const int i0 = part * 128;

  float a0 = 0.f, a1 = 0.f;
  const float* up = U + ((size_t)b * IN_CAPS + i0) * N_DIM;
  for (int ii = 0; ii < 128; ++ii) {
    const float cw = cij[i0 + ii];       // uniform -> scalar load
    a0 += up[t] * cw;
    a1 += up[t + 256] * cw;
    up += N_DIM;
  }
  float* sp = sjp + ((size_t)part * BATCH + b) * N_DIM;
  sp[t] = a0;
  sp[t + 256] = a1;
}

// ---------------------------------------------------------------------------
// reduce sj partials (fixed order), squash, write vj (and d_out on last iter)
// ---------------------------------------------------------------------------
__global__ __launch_bounds__(256)
void squash_vj(const float* __restrict__ sjp, float* __restrict__ vj,
               float* __restrict__ outp, int writeOut) {
  __shared__ float red[256];
  const int b = blockIdx.x;
  const int t = threadIdx.x;

  float s0 = 0.f, s1 = 0.f;
  for (int p = 0; p < SJ_SPLIT; ++p) {
    const float* sp = sjp + ((size_t)p * BATCH + b) * N_DIM;
    s0 += sp[t];
    s1 += sp[t + 256];
  }
  red[t] = s0 * s0 + s1 * s1;
  __syncthreads();
  for (int s = 128; s > 0; s >>= 1) {
    if (t < s) red[t] += red[t + s];
    __syncthreads();
  }
  const float nsq = red[0];
  const float norm = sqrtf(nsq);
  const float scale = norm / (1.f + nsq);   // ||s|| / (1 + ||s||^2)

  const float v0 = s0 * scale, v1 = s1 * scale;
  vj[(size_t)b * N_DIM + t] = v0;
  vj[(size_t)b * N_DIM + t + 256] = v1;
  if (writeOut) {
    outp[(size_t)b * N_DIM + t] = v0;
    outp[(size_t)b * N_DIM + t + 256] = v1;
  }
}

// ---------------------------------------------------------------------------
// routing update: c[i] += sum_{b,o} u[b,i,o] * vj[b,o]   (one block per i)
// ---------------------------------------------------------------------------
__global__ __launch_bounds__(256)
void update_c(const float* __restrict__ U, const float* __restrict__ vj,
              float* __restrict__ c) {
  __shared__ float red[256];
  const int i = blockIdx.x;
  const int t = threadIdx.x;

  float acc = 0.f;
  for (int b = 0; b < BATCH; ++b) {
    const float* up = U + ((size_t)b * IN_CAPS + i) * N_DIM;
    const float* vp = vj + (size_t)b * N_DIM;
    acc += up[t] * vp[t] + up[t + 256] * vp[t + 256];
  }
  red[t] = acc;
  __syncthreads();
  for (int s = 128; s > 0; s >>= 1) {
    if (t < s) red[t] += red[t + s];
    __syncthreads();
  }
  if (t == 0) c[i] += red[0];
}

// ---------------------------------------------------------------------------
extern "C" void kernel_launch(void* const* d_in, const int* in_sizes, int n_in,
                              void* d_out, int out_size, void* d_ws,
                              size_t ws_size, hipStream_t stream) {
  const float* input_x = (const float*)d_in[0];  // [64,1152,512]
  const float* W       = (const float*)d_in[1];  // [512,512]
  const float* bias    = (const float*)d_in[2];  // [512]
  const float* coeffs  = (const float*)d_in[3];  // [1,1152,1]
  float* out = (float*)d_out;                    // [64,512]

  // Workspace layout (floats)
  float* u   = (float*)d_ws;                          // 73728*512
  float* c   = u + (size_t)M_DIM * N_DIM;             // 1152
  float* cij = c + IN_CAPS;                           // 1152
  float* sjp = cij + IN_CAPS;                         // 9*64*512
  float* vj  = sjp + (size_t)SJ_SPLIT * BATCH * N_DIM;// 64*512
  float* Wt  = vj + (size_t)BATCH * N_DIM;            // 512*512

  init_c<<<(IN_CAPS + 255) / 256, 256, 0, stream>>>(coeffs, c);
  transpose_w<<<dim3(K_DIM / 32, N_DIM / 32), 256, 0, stream>>>(W, Wt);

  caps_gemm_wmma<<<dim3(N_DIM / 64, M_DIM / 64), 32, 0, stream>>>(
      input_x, Wt, bias, u);

  for (int it = 0; it < N_ITERS; ++it) {
    softmax_c<<<1, 256, 0, stream>>>(c, cij);
    sj_partial<<<dim3(BATCH, SJ_SPLIT), 256, 0, stream>>>(u, cij, sjp);
    squash_vj<<<BATCH, 256, 0, stream>>>(sjp, vj, out,
                                         (it == N_ITERS - 1) ? 1 : 0);
    if (it < N_ITERS - 1)
      update_c<<<IN_CAPS, 256, 0, stream>>>(u, vj, c);
  }
}